// SparseGATLayer_34187939676736
// MI455X (gfx1250) — compile-verified
//
#include <hip/hip_runtime.h>
#include <math.h>

// ---------------------------------------------------------------------------
// SparseGAT layer for MI455X (gfx1250, wave32).
//   h:(1,N,128) f32, Wq/Wk/Wv:(128,128) f32, aq/ak:(4,32) f32, edge_index:(2,L) i64
// Key algebra: q,k only appear via dot with aq/ak -> fold into 128x4 matrices.
// Real GEMM: V = h @ Wv done with V_WMMA_F32_16X16X4_F32 (full f32 precision).
// ---------------------------------------------------------------------------

typedef float v2f __attribute__((ext_vector_type(2)));
typedef float v8f __attribute__((ext_vector_type(8)));

#define IN_C 128
#define NHEAD 4
#define HD 32

// ---- monotone float<->uint mapping for atomicMax on floats -----------------
__device__ __forceinline__ unsigned fkey(float f) {
  unsigned u = __float_as_uint(f);
  return (u & 0x80000000u) ? ~u : (u | 0x80000000u);
}
__device__ __forceinline__ float fdec(unsigned k) {
  unsigned u = (k & 0x80000000u) ? (k & 0x7FFFFFFFu) : ~k;
  return __uint_as_float(u);
}

// ---- 1) fold Wq with aq, Wk with ak: Aq/Ak stored head-major [h*128 + c] ----
__global__ void fold_kernel(const float* __restrict__ Wq, const float* __restrict__ Wk,
                            const float* __restrict__ aq, const float* __restrict__ ak,
                            float* __restrict__ Aq, float* __restrict__ Ak) {
  int c = threadIdx.x;  // 0..127
  if (c >= IN_C) return;
  for (int hh = 0; hh < NHEAD; ++hh) {
    float sq = 0.f, sk = 0.f;
    for (int d = 0; d < HD; ++d) {
      sq += Wq[c * IN_C + hh * HD + d] * aq[hh * HD + d];
      sk += Wk[c * IN_C + hh * HD + d] * ak[hh * HD + d];
    }
    Aq[hh * IN_C + c] = sq;
    Ak[hh * IN_C + c] = sk;
  }
}

// ---- 2) V = h @ Wv via f32 WMMA (16x16 tile per wave, K stepped by 4) ------
// A 16x4 f32 layout (ISA 7.12.2): lanes 0-15 M=0..15 ; VGPR0 = K{0|2}, VGPR1 = K{1|3}
// B 4x16 mirrored across lanes (N = lane&15). C/D: VGPR r -> M = r (+8 for hi lanes).
__global__ void __launch_bounds__(256) gemm_v_kernel(const float* __restrict__ hsrc,
                                                     const float* __restrict__ Wv,
                                                     float* __restrict__ V, int N) {
  __shared__ float As[16 * IN_C];  // 8 KB: 16 rows of h shared by all 8 waves
  const int rowBase = blockIdx.x * 16;
  for (int i = threadIdx.x; i < 16 * IN_C; i += 256)
    As[i] = hsrc[rowBase * IN_C + i];
  __syncthreads();

  const int wave    = threadIdx.x >> 5;   // 8 waves -> 8 column tiles of 16
  const int lane    = threadIdx.x & 31;
  const int colBase = wave * 16;
  const int m       = lane & 15;
  const int kh      = (lane >> 4) << 1;   // 0 for lanes 0-15, 2 for lanes 16-31

  v8f acc = {};
#pragma unroll 8
  for (int kk = 0; kk < IN_C; kk += 4) {
    v2f a, b;
    a.x = As[m * IN_C + kk + kh];
    a.y = As[m * IN_C + kk + kh + 1];
    b.x = Wv[(kk + kh) * IN_C + colBase + m];
    b.y = Wv[(kk + kh + 1) * IN_C + colBase + m];
    acc = __builtin_amdgcn_wmma_f32_16x16x4_f32(false, a, false, b,
                                                (short)0, acc, false, false);
  }

  const int rOff = (lane >> 4) << 3;      // hi lane half holds M = r+8
  if (rowBase + 16 <= N) {
    // uniform fast path: no per-row EXEC guarding
#pragma unroll
    for (int r = 0; r < 8; ++r)
      V[(rowBase + rOff + r) * IN_C + colBase + m] = acc[r];
  } else {
    // ragged tail (never taken for N % 16 == 0)
#pragma unroll
    for (int r = 0; r < 8; ++r) {
      int row = rowBase + rOff + r;
      if (row < N) V[row * IN_C + colBase + m] = acc[r];
    }
  }
}

// ---- 3) sq = h @ Aq, sk = h @ Ak : one thread per (node, head) -------------
__global__ void sqsk_kernel(const float* __restrict__ hsrc,
                            const float* __restrict__ Aq, const float* __restrict__ Ak,
                            float* __restrict__ sq, float* __restrict__ sk, int N) {
  int idx = blockIdx.x * blockDim.x + threadIdx.x;
  if (idx >= N * NHEAD) return;
  int n = idx >> 2, hh = idx & 3;
  float aq_ = 0.f, ak_ = 0.f;
  const float* hr  = hsrc + n * IN_C;
  const float* aqr = Aq + hh * IN_C;
  const float* akr = Ak + hh * IN_C;
#pragma unroll 4
  for (int c = 0; c < IN_C; ++c) {
    float hv = hr[c];
    aq_ += hv * aqr[c];
    ak_ += hv * akr[c];
  }
  sq[n * NHEAD + hh] = aq_;
  sk[n * NHEAD + hh] = ak_;
}

// ---- 4) init: zero out, den; mx = 0 (== minimal fkey) ----------------------
__global__ void init_kernel(float* __restrict__ out, float* __restrict__ den,
                            unsigned* __restrict__ mx, int N) {
  int idx = blockIdx.x * blockDim.x + threadIdx.x;
  if (idx < N * IN_C) out[idx] = 0.f;
  if (idx < N * NHEAD) { den[idx] = 0.f; mx[idx] = 0u; }
}

// ---- 5) per-edge segment max of leaky_relu(sq[recv]+sk[send]) --------------
__global__ void edge_max_kernel(const long long* __restrict__ ei,
                                const float* __restrict__ sq, const float* __restrict__ sk,
                                unsigned* __restrict__ mx, int L) {
  int e = blockIdx.x * blockDim.x + threadIdx.x;
  if (e >= L) return;
  int r = (int)ei[e];
  int s = (int)ei[L + e];
#pragma unroll
  for (int hh = 0; hh < NHEAD; ++hh) {
    float sc = sq[r * NHEAD + hh] + sk[s * NHEAD + hh];
    sc = sc > 0.f ? sc : 0.2f * sc;  // leaky_relu(0.2)
    atomicMax(&mx[r * NHEAD + hh], fkey(sc));
  }
}

// ---- 6) per-edge exp / segment sum -----------------------------------------
__global__ void edge_sum_kernel(const long long* __restrict__ ei,
                                const float* __restrict__ sq, const float* __restrict__ sk,
                                const unsigned* __restrict__ mx,
                                float* __restrict__ den, int L) {
  int e = blockIdx.x * blockDim.x + threadIdx.x;
  if (e >= L) return;
  int r = (int)ei[e];
  int s = (int)ei[L + e];
#pragma unroll
  for (int hh = 0; hh < NHEAD; ++hh) {
    float sc = sq[r * NHEAD + hh] + sk[s * NHEAD + hh];
    sc = sc > 0.f ? sc : 0.2f * sc;
    float ex = __expf(sc - fdec(mx[r * NHEAD + hh]));
    atomicAdd(&den[r * NHEAD + hh], ex);
  }
}

// ---- 7) weighted scatter: one wave32 per edge, lane = d index --------------
// Coalesced 128B reads of v[send] per head; coalesced atomic adds to out[recv].
__global__ void __launch_bounds__(256) scatter_kernel(
    const long long* __restrict__ ei,
    const float* __restrict__ sq, const float* __restrict__ sk,
    const unsigned* __restrict__ mx, const float* __restrict__ den,
    const float* __restrict__ V, float* __restrict__ out, int L) {
  int e = blockIdx.x * 8 + (threadIdx.x >> 5);
  if (e >= L) return;
  int lane = threadIdx.x & 31;  // d index within a head
  int r = (int)ei[e];
  int s = (int)ei[L + e];
#pragma unroll
  for (int hh = 0; hh < NHEAD; ++hh) {
    float sc = sq[r * NHEAD + hh] + sk[s * NHEAD + hh];
    sc = sc > 0.f ? sc : 0.2f * sc;
    float ex = __expf(sc - fdec(mx[r * NHEAD + hh]));
    float d  = den[r * NHEAD + hh];
    float alpha = ex / fmaxf(d, 1e-12f);
    float vv = V[s * IN_C + hh * HD + lane];
    atomicAdd(&out[r * IN_C + hh * HD + lane], alpha * vv);
  }
}

// ---- 8) ELU in place --------------------------------------------------------
__global__ void elu_kernel(float* __restrict__ out, int n) {
  int idx = blockIdx.x * blockDim.x + threadIdx.x;
  if (idx >= n) return;
  float x = out[idx];
  out[idx] = x > 0.f ? x : (expf(x) - 1.0f);
}

// ---------------------------------------------------------------------------
extern "C" void kernel_launch(void* const* d_in, const int* in_sizes, int n_in,
                              void* d_out, int out_size, void* d_ws, size_t ws_size,
                              hipStream_t stream) {
  const float*     h  = (const float*)d_in[0];
  const float*     Wq = (const float*)d_in[1];
  const float*     Wk = (const float*)d_in[2];
  const float*     Wv = (const float*)d_in[3];
  const float*     aq = (const float*)d_in[4];
  const float*     ak = (const float*)d_in[5];
  const long long* ei = (const long long*)d_in[6];

  const int N = in_sizes[0] / IN_C;   // 50000
  const int L = in_sizes[6] / 2;      // 1600000

  // workspace layout (floats)
  float* ws  = (float*)d_ws;
  float* Aq  = ws;                    // 4*128
  float* Ak  = Aq + NHEAD * IN_C;     // 4*128
  float* sq  = Ak + NHEAD * IN_C;     // N*4
  float* sk  = sq + (size_t)N * NHEAD;
  unsigned* mx = (unsigned*)(sk + (size_t)N * NHEAD);  // N*4
  float* den = (float*)(mx + (size_t)N * NHEAD);       // N*4
  float* V   = den + (size_t)N * NHEAD;                // N*128
  float* out = (float*)d_out;

  // 1) fold attention vectors into weights
  fold_kernel<<<1, 128, 0, stream>>>(Wq, Wk, aq, ak, Aq, Ak);

  // 2) V = h @ Wv  (WMMA f32 16x16x4; 16 rows per block, 8 col-tiles = 8 waves)
  gemm_v_kernel<<<(N + 15) / 16, 256, 0, stream>>>(h, Wv, V, N);

  // 3) per-node attention scalars
  sqsk_kernel<<<(N * NHEAD + 255) / 256, 256, 0, stream>>>(h, Aq, Ak, sq, sk, N);

  // 4) init accumulators
  init_kernel<<<(N * IN_C + 255) / 256, 256, 0, stream>>>(out, den, mx, N);

  // 5-6) segment softmax over incoming edges
  edge_max_kernel<<<(L + 255) / 256, 256, 0, stream>>>(ei, sq, sk, mx, L);
  edge_sum_kernel<<<(L + 255) / 256, 256, 0, stream>>>(ei, sq, sk, mx, den, L);

  // 7) weighted message aggregation (wave per edge)
  scatter_kernel<<<(L + 7) / 8, 256, 0, stream>>>(ei, sq, sk, mx, den, V, out, L);

  // 8) ELU
  elu_kernel<<<(N * IN_C + 255) / 256, 256, 0, stream>>>(out, N * IN_C);
}